// Critic_403726926482
// MI455X (gfx1250) — compile-verified
//
#include <hip/hip_runtime.h>

typedef __attribute__((ext_vector_type(2))) float v2f;
typedef __attribute__((ext_vector_type(8))) float v8f;
typedef __attribute__((ext_vector_type(4))) unsigned int u32x4;
typedef __attribute__((ext_vector_type(4))) int i32x4;
typedef __attribute__((ext_vector_type(8))) int i32x8;

#define FIN 256
#define FH  48
#define FC  16

// ---------------- TDM: stage a whole row-major 2D fp32 tensor into LDS ----------------
// dim0 = contiguous length (elements), dim1 = number of rows.
// D# layout per CDNA5 ISA sec 8.3-8.5: group0 {count=1, lds_addr, global_addr, type=2},
// group1 {data_size=4B, tensor dims, tile dims = whole tensor, dim0_stride = dim0}.
__device__ __forceinline__ void tdm_load_2d_f32(float* lds_ptr, const float* gptr,
                                                unsigned dim0, unsigned dim1) {
  unsigned lds = (unsigned)(unsigned long long)lds_ptr;   // flat low 32 bits == LDS byte addr
  unsigned long long ga = (unsigned long long)gptr;

  u32x4 g0;
  g0.x = 1u;                                  // count=1 (valid user descriptor)
  g0.y = lds;                                 // lds_addr
  g0.z = (unsigned)ga;                        // global_addr[31:0]
  g0.w = (unsigned)(ga >> 32) | (2u << 30);   // global_addr[56:32] | type=2 ("image")

  i32x8 g1;
  g1[0] = (int)(2u << 16);                            // data_size=2 -> 4 bytes
  g1[1] = (int)((dim0 & 0xffffu) << 16);              // tensor_dim0[15:0]
  g1[2] = (int)(((dim0 >> 16) & 0xffffu) | ((dim1 & 0xffffu) << 16));  // dim0 hi | dim1 lo
  g1[3] = (int)(((dim1 >> 16) & 0xffffu) | ((dim0 & 0xffffu) << 16));  // dim1 hi | tile_dim0
  g1[4] = (int)(dim1 & 0xffffu);                      // tile_dim1 (tile_dim2=0)
  g1[5] = (int)dim0;                                  // tensor_dim0_stride[31:0]
  g1[6] = 0;                                          // stride hi | tensor_dim1_stride lo
  g1[7] = 0;

  i32x4 gz = {};
#if __clang_major__ >= 23
  i32x8 gz8 = {};
  __builtin_amdgcn_tensor_load_to_lds(g0, g1, gz, gz, gz8, 0);
#else
  __builtin_amdgcn_tensor_load_to_lds(g0, g1, gz, gz, 0);
#endif
}

// ---------------- utility kernels ----------------

__global__ void fill_f32(float* __restrict__ p, float v, int n) {
  int i = blockIdx.x * blockDim.x + threadIdx.x;
  if (i < n) p[i] = v;
}

__global__ void degree_kernel(const int* __restrict__ dst, float* __restrict__ deg, int E) {
  int e = blockIdx.x * blockDim.x + threadIdx.x;
  if (e < E) atomicAdd(&deg[dst[e]], 1.0f);
}

__global__ void rsqrt_kernel(float* __restrict__ deg, int n) {
  int i = blockIdx.x * blockDim.x + threadIdx.x;
  if (i < n) deg[i] = rsqrtf(deg[i]);
}

// ---------------- GEMM1: H1[N,48] = X[N,256] @ W1[256,48] (fp32 WMMA) ----------------
// 128 threads = 4 waves; each wave computes a 16x48 tile (3 accumulators).
// W1 staged via Tensor Data Mover (TENSORcnt), issued by wave 0.

__global__ __launch_bounds__(128)
void gemm1_wmma(const float* __restrict__ X, const float* __restrict__ W,
                float* __restrict__ H, int n) {
  __shared__ float sW[FIN * FH];  // 48 KB
  const int tid  = threadIdx.x;
  const int wave = tid >> 5;
  const int lane = tid & 31;

  if (tid == 0) {
    tdm_load_2d_f32(sW, W, FH, FIN);            // contig=48, rows=256
    __builtin_amdgcn_s_wait_tensorcnt(0);
  }
  __syncthreads();

  const int rowBase = blockIdx.x * 64 + wave * 16;
  const int row  = min(rowBase + (lane & 15), n - 1);   // clamp for tail (loads only)
  const int klo  = (lane >> 4) << 1;                    // 0 or 2
  const int col  = lane & 15;
  const float* xr = X + (long long)row * FIN;

  v8f acc0 = {}, acc1 = {}, acc2 = {};
  for (int k = 0; k < FIN; k += 4) {
    v2f a; a.x = xr[k + klo]; a.y = xr[k + klo + 1];
    const float* w0 = &sW[(k + klo) * FH];
    const float* w1 = w0 + FH;
    v2f b0; b0.x = w0[col];      b0.y = w1[col];
    v2f b1; b1.x = w0[col + 16]; b1.y = w1[col + 16];
    v2f b2; b2.x = w0[col + 32]; b2.y = w1[col + 32];
    acc0 = __builtin_amdgcn_wmma_f32_16x16x4_f32(false, a, false, b0, (short)0, acc0, false, false);
    acc1 = __builtin_amdgcn_wmma_f32_16x16x4_f32(false, a, false, b1, (short)0, acc1, false, false);
    acc2 = __builtin_amdgcn_wmma_f32_16x16x4_f32(false, a, false, b2, (short)0, acc2, false, false);
  }

  const int wrow = rowBase + ((lane >> 4) << 3);
  for (int j = 0; j < 8; ++j) {
    int gr = wrow + j;
    if (gr < n) {
      float* o = H + (long long)gr * FH + col;
      o[0]  = acc0[j];
      o[16] = acc1[j];
      o[32] = acc2[j];
    }
  }
}

// ---------------- GEMM2: H2[N,16] = X1[N,48] @ W2[48,16] (fp32 WMMA) ----------------

__global__ __launch_bounds__(256)
void gemm2_wmma(const float* __restrict__ X, const float* __restrict__ W,
                float* __restrict__ H, int n) {
  __shared__ float sW[FH * FC];  // 3 KB
  const int tid  = threadIdx.x;
  const int wave = tid >> 5;
  const int lane = tid & 31;

  if (tid == 0) {
    tdm_load_2d_f32(sW, W, FC, FH);             // contig=16, rows=48
    __builtin_amdgcn_s_wait_tensorcnt(0);
  }
  __syncthreads();

  const int rowBase = blockIdx.x * 128 + wave * 16;
  const int row  = min(rowBase + (lane & 15), n - 1);
  const int klo  = (lane >> 4) << 1;
  const int col  = lane & 15;
  const float* xr = X + (long long)row * FH;

  v8f acc = {};
  for (int k = 0; k < FH; k += 4) {
    v2f a; a.x = xr[k + klo]; a.y = xr[k + klo + 1];
    v2f b; b.x = sW[(k + klo) * FC + col]; b.y = sW[(k + klo + 1) * FC + col];
    acc = __builtin_amdgcn_wmma_f32_16x16x4_f32(false, a, false, b, (short)0, acc, false, false);
  }

  const int wrow = rowBase + ((lane >> 4) << 3);
  for (int j = 0; j < 8; ++j) {
    int gr = wrow + j;
    if (gr < n) H[(long long)gr * FC + col] = acc[j];
  }
}

// ---------------- Edge aggregation (scatter-add with sym norm) ----------------
// One thread per (edge, 4-float chunk). Layer1: 12 chunks, Layer2: 4 chunks.

__global__ void agg1_kernel(const int* __restrict__ src, const int* __restrict__ dst,
                            const float* __restrict__ dinv, const float* __restrict__ H1,
                            float* __restrict__ out, int E) {
  int gid = blockIdx.x * blockDim.x + threadIdx.x;
  int e = gid / 12;
  if (e >= E) return;
  int c = (gid - e * 12) * 4;
  __builtin_prefetch(src + e + 4096, 0, 0);
  __builtin_prefetch(dst + e + 4096, 0, 0);
  int s = src[e], d = dst[e];
  float w = dinv[s] * dinv[d];
  float4 v = *(const float4*)(H1 + (long long)s * FH + c);
  float* o = out + (long long)d * FH + c;
  atomicAdd(o + 0, v.x * w);
  atomicAdd(o + 1, v.y * w);
  atomicAdd(o + 2, v.z * w);
  atomicAdd(o + 3, v.w * w);
}

__global__ void agg2_kernel(const int* __restrict__ src, const int* __restrict__ dst,
                            const float* __restrict__ dinv, const float* __restrict__ H2,
                            float* __restrict__ out, int E) {
  int gid = blockIdx.x * blockDim.x + threadIdx.x;
  int e = gid >> 2;
  if (e >= E) return;
  int c = (gid & 3) * 4;
  __builtin_prefetch(src + e + 4096, 0, 0);
  __builtin_prefetch(dst + e + 4096, 0, 0);
  int s = src[e], d = dst[e];
  float w = dinv[s] * dinv[d];
  float4 v = *(const float4*)(H2 + (long long)s * FC + c);
  float* o = out + (long long)d * FC + c;
  atomicAdd(o + 0, v.x * w);
  atomicAdd(o + 1, v.y * w);
  atomicAdd(o + 2, v.z * w);
  atomicAdd(o + 3, v.w * w);
}

// ---------------- self-loop + bias (+ relu for layer 1) ----------------

__global__ void selfloop_relu(const float* __restrict__ H1, const float* __restrict__ dinv,
                              const float* __restrict__ b1, float* __restrict__ x1, int n) {
  int gid = blockIdx.x * blockDim.x + threadIdx.x;
  if (gid >= n * FH) return;
  int i = gid / FH;
  int c = gid - i * FH;
  float di = dinv[i];
  float v = x1[gid] + H1[gid] * di * di + b1[c];
  x1[gid] = v > 0.0f ? v : 0.0f;
}

__global__ void final_kernel(const float* __restrict__ H2, const float* __restrict__ dinv,
                             const float* __restrict__ b2, float* __restrict__ out, int n) {
  int gid = blockIdx.x * blockDim.x + threadIdx.x;
  if (gid >= n * FC) return;
  int i = gid >> 4;
  int c = gid & 15;
  float di = dinv[i];
  out[gid] += H2[gid] * di * di + b2[c];
}

// ---------------- launch ----------------

extern "C" void kernel_launch(void* const* d_in, const int* in_sizes, int n_in,
                              void* d_out, int out_size, void* d_ws, size_t ws_size,
                              hipStream_t stream) {
  const int*   edge = (const int*)d_in[0];
  const float* X    = (const float*)d_in[1];
  const float* W1   = (const float*)d_in[2];
  const float* b1   = (const float*)d_in[3];
  const float* W2   = (const float*)d_in[4];
  const float* b2   = (const float*)d_in[5];
  float* out = (float*)d_out;

  const int E = in_sizes[0] / 2;
  const int N = in_sizes[1] / FIN;
  const int* src = edge;
  const int* dst = edge + E;

  // workspace layout (floats): dinv[N] | H1[N*FH] | A1[N*FH] | H2[N*FC]  (~45 MB)
  float* deg = (float*)d_ws;
  float* H1  = deg + N;
  float* A1  = H1 + (size_t)N * FH;
  float* H2  = A1 + (size_t)N * FH;

  const int TB = 256;
  fill_f32<<<(N + TB - 1) / TB, TB, 0, stream>>>(deg, 1.0f, N);          // self-loop degree
  fill_f32<<<(N * FH + TB - 1) / TB, TB, 0, stream>>>(A1, 0.0f, N * FH);
  fill_f32<<<(N * FC + TB - 1) / TB, TB, 0, stream>>>(out, 0.0f, N * FC);

  degree_kernel<<<(E + TB - 1) / TB, TB, 0, stream>>>(dst, deg, E);
  rsqrt_kernel<<<(N + TB - 1) / TB, TB, 0, stream>>>(deg, N);            // deg -> dinv in place

  gemm1_wmma<<<(N + 63) / 64, 128, 0, stream>>>(X, W1, H1, N);

  {
    long long t = (long long)E * 12;
    agg1_kernel<<<(int)((t + TB - 1) / TB), TB, 0, stream>>>(src, dst, deg, H1, A1, E);
  }
  selfloop_relu<<<(N * FH + TB - 1) / TB, TB, 0, stream>>>(H1, deg, b1, A1, N);

  gemm2_wmma<<<(N + 127) / 128, 256, 0, stream>>>(A1, W2, H2, N);

  {
    long long t = (long long)E * 4;
    agg2_kernel<<<(int)((t + TB - 1) / TB), TB, 0, stream>>>(src, dst, deg, H2, out, E);
  }
  final_kernel<<<(N * FC + TB - 1) / TB, TB, 0, stream>>>(H2, deg, b2, out, N);
}